// BinaryBNModel_5540507812483
// MI455X (gfx1250) — compile-verified
//
#include <hip/hip_runtime.h>

// CDNA5 / gfx1250, wave32. Matrix work uses V_WMMA_F32_16X16X4_F32 (full f32,
// matches the f32 reference numerics).
typedef float v2f __attribute__((ext_vector_type(2)));
typedef float v8f __attribute__((ext_vector_type(8)));

#define K_VARS 8     // K in reference
#define C_COMB 256   // 2^K

// One wave owns a 16-row batch tile and a strided subset of tables t.
// For each t:
//   pLo[m][k]  (vars 4..7, k=c_lo in 0..15)  -> WMMA A operand (built in-lane)
//   cpds_t[n*16+k]                            -> WMMA B operand (float2 loads)
//   Y = PLo x Cpds_t^T via 4x v_wmma_f32_16x16x4_f32 (K=16 chained)
//   out[m] += sum_n pHi[m][n] * Y[m][n]       (per-lane FMA, n = lane&15)
//
// NVC > 0: compile-time row stride -> x gathers use one 32-bit voffset per
// fv with the r*NV term folded into the 24-bit instruction offset (GVS mode).
template <int NVC>
__global__ __launch_bounds__(256) void bn_wmma_kernel(
    const float* __restrict__ x, const float* __restrict__ cpds,
    const int* __restrict__ fvars, float* __restrict__ dst,
    int B, int nv_rt, int T, int NS)
{
  const int NV = (NVC > 0) ? NVC : nv_rt;

  __shared__ float wred[8][16];

  const int lane = threadIdx.x & 31;
  const int wid  = __builtin_amdgcn_readfirstlane((int)(threadIdx.x >> 5));
  const int m    = lane & 15;   // tile row; equals WMMA column n for C/D layout
  const int hb   = lane >> 4;   // wave half selects K-subcolumns (koff = 2*hb)
  const int rowBase = blockIdx.x * 16;
  const int ts   = blockIdx.y;          // t-slice
  const int sub  = ts * 8 + wid;        // global sub-slice for this wave
  const int step = NS * 8;

  // bits of n = m for the pHi factors (i = 0..3 -> bit (3-j) of c_hi)
  const int bb0 = (m >> 3) & 1, bb1 = (m >> 2) & 1;
  const int bb2 = (m >> 1) & 1, bb3 = m & 1;

  float vout[8];
#pragma unroll
  for (int r = 0; r < 8; ++r) vout[r] = 0.f;

  // Uniform SGPR base for the batch tile; all gathers use 32-bit lane offsets.
  const float* __restrict__ xb = x + (size_t)rowBase * NV;
  const unsigned roff = (unsigned)(m * NV);       // own row (pLo vars)
  const unsigned hoff = (unsigned)(hb * 8 * NV);  // first row of this half (pHi)

  for (int t = sub; t < T; t += step) {
    // wave-uniform -> two s_load_b128
    const int4 fA = *(const int4*)(fvars + t * K_VARS);      // vars 0..3 (pHi)
    const int4 fB = *(const int4*)(fvars + t * K_VARS + 4);  // vars 4..7 (pLo)

    // ---- A operands: pLo[m][k] for k = 4q + 2*hb (+1) ----
    // pLo[m][k] = f0(k>>3) * f1((k>>2)&1) * f2((k>>1)&1) * f3(k&1),
    // with f_j(b) = b ? v[4+j] : 1-v[4+j]; k is even (x lane) / odd (y lane).
    const float v4 = xb[roff + (unsigned)fB.x];
    const float v5 = xb[roff + (unsigned)fB.y];
    const float v6 = xb[roff + (unsigned)fB.z];
    const float v7 = xb[roff + (unsigned)fB.w];
    const float u4 = 1.f - v4, u5 = 1.f - v5, u6 = 1.f - v6, u7 = 1.f - v7;
    const float c2  = hb ? v6 : u6;                  // bit2 of k equals hb
    const float e0  = u4 * c2, e1 = v4 * c2;
    const float t00 = e0 * u5, t01 = e0 * v5, t10 = e1 * u5, t11 = e1 * v5;
    const v2f A0 = { t00 * u7, t00 * v7 };           // q=0 (bits b0=0,b1=0)
    const v2f A1 = { t01 * u7, t01 * v7 };           // q=1
    const v2f A2 = { t10 * u7, t10 * v7 };           // q=2
    const v2f A3 = { t11 * u7, t11 * v7 };           // q=3

    // ---- B operands: cpds_t as B[k=c_lo][n=c_hi], float2 per lane ----
    // Uniform base + 32-bit lane offset; +4q folds into the imm offset.
    const float* __restrict__ cbase = cpds + (size_t)t * C_COMB;
    const unsigned coff = (unsigned)(m * 16 + 2 * hb);
    const v2f B0 = *(const v2f*)(cbase + coff);
    const v2f B1 = *(const v2f*)(cbase + coff + 4);
    const v2f B2 = *(const v2f*)(cbase + coff + 8);
    const v2f B3 = *(const v2f*)(cbase + coff + 12);

    if (t + step < T)  // stream next cpds tile (global_prefetch_b8)
      __builtin_prefetch(cbase + (size_t)step * C_COMB + coff, 0, 3);

    // ---- Y = PLo x Cpds_t^T : K=16 chained as 4x 16x16x4 f32 WMMA ----
    v8f acc = {};
    acc = __builtin_amdgcn_wmma_f32_16x16x4_f32(false, A0, false, B0, (short)0, acc, false, false);
    acc = __builtin_amdgcn_wmma_f32_16x16x4_f32(false, A1, false, B1, (short)0, acc, false, false);
    acc = __builtin_amdgcn_wmma_f32_16x16x4_f32(false, A2, false, B2, (short)0, acc, false, false);
    acc = __builtin_amdgcn_wmma_f32_16x16x4_f32(false, A3, false, B3, (short)0, acc, false, false);

    // ---- out[m_r] += sum_n pHi[m_r][n] * Y[m_r][n]; this lane covers n=m ----
    // acc[r] holds Y[r + 8*hb][n]. One 32-bit voffset per fv; the unrolled
    // r*NV term becomes the instruction's immediate byte offset.
    const unsigned i0 = hoff + (unsigned)fA.x;
    const unsigned i1 = hoff + (unsigned)fA.y;
    const unsigned i2 = hoff + (unsigned)fA.z;
    const unsigned i3 = hoff + (unsigned)fA.w;
#pragma unroll
    for (int r = 0; r < 8; ++r) {
      const unsigned ro = (unsigned)(r * NV);   // constant when NVC > 0
      const float a0 = xb[i0 + ro];
      const float a1 = xb[i1 + ro];
      const float a2 = xb[i2 + ro];
      const float a3 = xb[i3 + ro];
      const float g = (bb0 ? a0 : 1.f - a0) * (bb1 ? a1 : 1.f - a1)
                    * (bb2 ? a2 : 1.f - a2) * (bb3 ? a3 : 1.f - a3);
      vout[r] = fmaf(g, acc[r], vout[r]);
    }
  }

  // ---- reduce over n (16 lanes per half) with a fixed xor tree ----
#pragma unroll
  for (int r = 0; r < 8; ++r) {
    float v = vout[r];
    v += __shfl_xor(v, 1, 32);
    v += __shfl_xor(v, 2, 32);
    v += __shfl_xor(v, 4, 32);
    v += __shfl_xor(v, 8, 32);
    if (m == 0) wred[wid][hb * 8 + r] = v;   // lanes 0 and 16 publish
  }
  __syncthreads();

  // cross-wave sum (fixed order -> deterministic), one write per row
  if (threadIdx.x < 16) {
    float s = 0.f;
#pragma unroll
    for (int w = 0; w < 8; ++w) s += wred[w][threadIdx.x];
    dst[(size_t)ts * B + rowBase + threadIdx.x] = s;
  }
}

// Final fixed-order sum over the NS t-slice partials (no float atomics).
__global__ __launch_bounds__(256) void bn_reduce_kernel(
    const float* __restrict__ part, float* __restrict__ out, int B, int NS)
{
  const int i = blockIdx.x * blockDim.x + threadIdx.x;
  if (i >= B) return;
  float s = 0.f;
  for (int k = 0; k < NS; ++k) s += part[(size_t)k * B + i];
  out[i] = s;
}

extern "C" void kernel_launch(void* const* d_in, const int* in_sizes, int n_in,
                              void* d_out, int out_size, void* d_ws, size_t ws_size,
                              hipStream_t stream) {
  const float* x     = (const float*)d_in[0];
  const float* cpds  = (const float*)d_in[1];
  const int*   fvars = (const int*)d_in[2];

  const int B  = out_size;                 // 1024
  const int NV = in_sizes[0] / B;          // 1024
  const int T  = in_sizes[2] / K_VARS;     // 400

  // Split T across gridDim.y slices if workspace allows (deterministic
  // two-pass reduction); otherwise single slice writes d_out directly.
  int NS = 8;
  if (ws_size < (size_t)NS * B * sizeof(float)) NS = 1;
  float* dst = (NS > 1) ? (float*)d_ws : (float*)d_out;

  dim3 grid(B / 16, NS);
  if (NV == 1024)
    bn_wmma_kernel<1024><<<grid, 256, 0, stream>>>(x, cpds, fvars, dst, B, NV, T, NS);
  else
    bn_wmma_kernel<0><<<grid, 256, 0, stream>>>(x, cpds, fvars, dst, B, NV, T, NS);

  if (NS > 1)
    bn_reduce_kernel<<<(B + 255) / 256, 256, 0, stream>>>((const float*)d_ws,
                                                          (float*)d_out, B, NS);
}